// SparseMiddleExtractor_1425929142951
// MI455X (gfx1250) — compile-verified
//
#include <hip/hip_runtime.h>
#include <hip/hip_bf16.h>

// ---------------- problem constants ----------------
#define BB   2
#define D0C  41
#define D1C  20        // (41-3)/2+1
#define D2C  9         // (20-3)/2+1
#define HH   200
#define WW   176
#define CIN  128
#define CM   64
#define NV0  30000     // B * N_PER
#define CAP1 60000     // <= 2*NV0 distinct outputs of conv_z 1
#define CAP2 120000    // <= 2*CAP1
#define M0C  (BB*D0C*HH*WW)   // 2,886,400
#define M1C  (BB*D1C*HH*WW)   // 1,408,000
#define M2C  (BB*D2C*HH*WW)   //   633,600
#define EPSF 1e-3f

typedef __attribute__((ext_vector_type(2))) float v2f;
typedef __attribute__((ext_vector_type(8))) float v8f;

#if __has_builtin(__builtin_amdgcn_global_load_async_to_lds_b128)
#define HAVE_ASYNC_LDS 1
typedef int v4i_b128 __attribute__((vector_size(16)));
typedef v4i_b128 __attribute__((address_space(1)))* async_gptr_t;  // global
typedef v4i_b128 __attribute__((address_space(3)))* async_lptr_t;  // LDS
#else
#define HAVE_ASYNC_LDS 0
#endif

static __device__ __forceinline__ void wait_async0() {
#if __has_builtin(__builtin_amdgcn_s_wait_asynccnt)
    __builtin_amdgcn_s_wait_asynccnt(0);
#else
    asm volatile("s_wait_asynccnt 0x0" ::: "memory");
#endif
}

// ---------------- tiny utility kernels ----------------
__global__ void fill_i32(int* __restrict__ p, int v, int n) {
    int i = blockIdx.x * 256 + threadIdx.x;
    if (i < n) p[i] = v;
}
__global__ void fill_f32(float* __restrict__ p, float v, int n) {
    int i = blockIdx.x * 256 + threadIdx.x;
    if (i < n) p[i] = v;
}
__global__ void init_cnts(int* __restrict__ cnts, int n0) {
    if (threadIdx.x == 0) cnts[0] = n0;
}

// weight transpose: [taps][K][64] -> [taps][64][stride] (stride = K+2, pad pre-zeroed)
__global__ void transpose_w(const float* __restrict__ w, float* __restrict__ wT,
                            int taps, int K, int stride) {
    int i = blockIdx.x * 256 + threadIdx.x;
    int total = taps * K * 64;
    if (i >= total) return;
    int n = i % 64;
    int k = (i / 64) % K;
    int t = i / (64 * K);
    wT[((size_t)t * 64 + n) * stride + k] = w[i];
}

// idx_map[lin(coords[i])] = i    (level-0 build; all rows valid)
__global__ void scatter_idx0(const int* __restrict__ coords, int n,
                             int* __restrict__ idx) {
    int i = blockIdx.x * 256 + threadIdx.x;
    if (i >= n) return;
    int b = coords[i*4+0], z = coords[i*4+1], y = coords[i*4+2], x = coords[i*4+3];
    idx[((b*D0C + z)*HH + y)*WW + x] = i;
}

// mark output occupancy for strided-z conv (benign write races: all write 1)
__global__ void mark_occ(const int* __restrict__ coords,
                         const int* __restrict__ cnts, int ci,
                         int* __restrict__ occ, int Dout) {
    int i = blockIdx.x * 256 + threadIdx.x;
    if (i >= cnts[ci]) return;
    int b = coords[i*4+0], z = coords[i*4+1], y = coords[i*4+2], x = coords[i*4+3];
    #pragma unroll
    for (int kz = 0; kz < 3; ++kz) {
        int t = z - kz;
        if (t >= 0 && (t & 1) == 0) {
            int oz = t >> 1;
            if (oz < Dout) occ[((b*Dout + oz)*HH + y)*WW + x] = 1;
        }
    }
}

// ---------------- deterministic prefix-scan compaction ----------------
__global__ void scan_local(const int* __restrict__ occ, int* __restrict__ local,
                           int* __restrict__ bsum, int M) {
    __shared__ int lds[256];
    int tid = threadIdx.x;
    int i = blockIdx.x * 256 + tid;
    int v = (i < M) ? occ[i] : 0;
    lds[tid] = v;
    __syncthreads();
    for (int off = 1; off < 256; off <<= 1) {
        int t = (tid >= off) ? lds[tid - off] : 0;
        __syncthreads();
        lds[tid] += t;
        __syncthreads();
    }
    if (i < M) local[i] = lds[tid] - v;          // exclusive prefix within block
    if (tid == 255) bsum[blockIdx.x] = lds[255]; // block total
}

__global__ void scan_blocksums(int* __restrict__ bs, int nb, int* __restrict__ totalOut) {
    __shared__ int lds[256];
    __shared__ int carry;
    int tid = threadIdx.x;
    if (tid == 0) carry = 0;
    __syncthreads();
    for (int base = 0; base < nb; base += 256) {
        int i = base + tid;
        int v = (i < nb) ? bs[i] : 0;
        lds[tid] = v;
        __syncthreads();
        for (int off = 1; off < 256; off <<= 1) {
            int t = (tid >= off) ? lds[tid - off] : 0;
            __syncthreads();
            lds[tid] += t;
            __syncthreads();
        }
        int incl = lds[tid];
        if (i < nb) bs[i] = carry + incl - v;    // exclusive + carry
        __syncthreads();
        if (tid == 255) carry += lds[255];
        __syncthreads();
    }
    if (tid == 0) *totalOut = carry;
}

// finalize: idx_map = compact row (or -1); emit coords of occupied cells
__global__ void scan_finalize(const int* __restrict__ occ, int* __restrict__ idxmap,
                              const int* __restrict__ bsum, int* __restrict__ coordsOut,
                              int Dout, int M) {
    int i = blockIdx.x * 256 + threadIdx.x;
    if (i >= M) return;
    if (occ[i]) {
        int row = bsum[blockIdx.x] + idxmap[i];
        idxmap[i] = row;
        const int HWc = HH * WW;
        const int DHW = Dout * HWc;
        int b = i / DHW;
        int r = i % DHW;
        int z = r / HWc;
        int y = (r % HWc) / WW;
        int x = r % WW;
        coordsOut[row*4+0] = b;
        coordsOut[row*4+1] = z;
        coordsOut[row*4+2] = y;
        coordsOut[row*4+3] = x;
    } else {
        idxmap[i] = -1;
    }
}

// ---------------- WMMA gather-GEMM conv ----------------
// MODE 0: submanifold 3x3x3 (27 taps, neighbor offsets via idx_map of SAME level)
// MODE 1: strided-z conv (3 taps, gather z_in = 2*z_out + kz via idx_map of INPUT level)
// Wave computes 16 rows x 64 cols; block (128 thr = 4 waves) covers 64 rows.
// Per-tap weight panel staged [n][K+2] in LDS via async global->LDS copies
// (overlapped with the neighbor-index gather), B fragments are single ds_load_b64.
template <int K, int MODE>
__global__ __launch_bounds__(128)
void conv_wmma(const float* __restrict__ feats,   // [n_in][K]
               const int*   __restrict__ coords,  // [cap][4] rows of the M dimension
               const int*   __restrict__ idx_map, // dense idx map used for gather
               const float* __restrict__ WtT,     // [taps][64][K+2] transposed weights
               float*       __restrict__ out,     // [cap][64]
               const int*   __restrict__ cnts, int ci,
               int Dmap) {
    constexpr int KS = K + 2;                     // padded row stride (floats)
    constexpr int PANEL = 64 * KS * 4;            // panel bytes
    __shared__ __align__(16) float Bs[64 * KS];
    const int n = cnts[ci];
    const int rowBase = blockIdx.x * 64;
    if (rowBase >= n) return;

    const int tid  = threadIdx.x;
    const int lane = tid & 31;
    const int wave = tid >> 5;
    const int myRowBase = rowBase + wave * 16;
    const int m = myRowBase + (lane & 15);
    const bool rowOK = (m < n);

    int cb = 0, cz = 0, cy = 0, cx = 0;
    if (rowOK) {
        cb = coords[m*4+0]; cz = coords[m*4+1]; cy = coords[m*4+2]; cx = coords[m*4+3];
    }

    v8f acc[4];
    #pragma unroll
    for (int nt = 0; nt < 4; ++nt)
        #pragma unroll
        for (int r = 0; r < 8; ++r) acc[nt][r] = 0.0f;

    const int khalf = 2 * (lane >> 4);  // 0 for lanes 0-15, 2 for lanes 16-31
    const int col   = lane & 15;
    const int TAPS  = (MODE == 0) ? 27 : 3;

    for (int t = 0; t < TAPS; ++t) {
        // -- stage this tap's weight panel into LDS (async, overlapped with gather) --
        {
            const char* gpanel = (const char*)WtT + (size_t)t * PANEL;
            char* lbase = (char*)Bs;
            for (int ofs = tid * 16; ofs < PANEL; ofs += 128 * 16) {
#if HAVE_ASYNC_LDS
                __builtin_amdgcn_global_load_async_to_lds_b128(
                    (async_gptr_t)(gpanel + ofs),
                    (async_lptr_t)(lbase + ofs), 0, 0);
#else
                *(float4*)(lbase + ofs) = *(const float4*)(gpanel + ofs);
#endif
            }
        }

        // -- neighbor gather index for this lane's row (overlaps the async copy) --
        int j = -1;
        if (rowOK) {
            if (MODE == 0) {
                int dz = t / 9 - 1, dy = (t / 3) % 3 - 1, dx = t % 3 - 1;
                int nz = cz + dz, ny = cy + dy, nx = cx + dx;
                if (nz >= 0 && nz < Dmap && ny >= 0 && ny < HH && nx >= 0 && nx < WW)
                    j = idx_map[((cb * Dmap + nz) * HH + ny) * WW + nx];
            } else {
                int zin = 2 * cz + t;
                if (zin < Dmap)
                    j = idx_map[((cb * Dmap + zin) * HH + cy) * WW + cx];
            }
        }

#if HAVE_ASYNC_LDS
        wait_async0();
#endif
        __syncthreads();

        const float* arow = feats + (size_t)(j >= 0 ? j : 0) * K;
        #pragma unroll 4
        for (int kk = 0; kk < K; kk += 4) {
            v2f a;
            if (j >= 0) {                       // contiguous 8B -> global_load_b64
                a[0] = arow[kk + khalf];
                a[1] = arow[kk + khalf + 1];
            } else {
                a[0] = 0.0f; a[1] = 0.0f;
            }
            #pragma unroll
            for (int nt = 0; nt < 4; ++nt) {
                // single aligned ds_load_b64, bank-conflict-free across N-lanes
                const v2f b = *(const v2f*)&Bs[(nt * 16 + col) * KS + kk + khalf];
                acc[nt] = __builtin_amdgcn_wmma_f32_16x16x4_f32(
                    false, a, false, b, (short)0, acc[nt], false, false);
            }
        }
        __syncthreads();
    }

    // store: D layout -> VGPR r holds M=r (lanes 0-15) / M=8+r (lanes 16-31), lane&15 = N
    #pragma unroll
    for (int nt = 0; nt < 4; ++nt)
        #pragma unroll
        for (int r = 0; r < 8; ++r) {
            int row = myRowBase + r + 8 * (lane >> 4);
            if (row < n)
                out[(size_t)row * 64 + nt * 16 + col] = acc[nt][r];
        }
}

// ---------------- batch-norm (+ReLU), deterministic ----------------
__global__ void bn_partial(const float* __restrict__ x, const int* __restrict__ cnts,
                           int ci, float* __restrict__ partial) {
    int c = threadIdx.x;               // 64 threads
    int n = cnts[ci];
    float s = 0.f, q = 0.f;
    for (int r = blockIdx.x; r < n; r += gridDim.x) {
        float v = x[(size_t)r * 64 + c];
        s += v;
        q += v * v;
    }
    partial[blockIdx.x * 128 + c]      = s;
    partial[blockIdx.x * 128 + 64 + c] = q;
}

__global__ void bn_finalize(const float* __restrict__ partial,
                            const float* __restrict__ gamma,
                            const float* __restrict__ beta,
                            const int* __restrict__ cnts, int ci,
                            float* __restrict__ coef, int G) {
    int c = threadIdx.x;               // 64 threads
    double s = 0.0, q = 0.0;
    for (int g = 0; g < G; ++g) {      // fixed order -> deterministic
        s += (double)partial[g * 128 + c];
        q += (double)partial[g * 128 + 64 + c];
    }
    int nv = cnts[ci]; if (nv < 1) nv = 1;
    double inv = 1.0 / (double)nv;
    double mean = s * inv;
    double var  = q * inv - mean * mean;
    if (var < 0.0) var = 0.0;
    float scale = gamma[c] * rsqrtf((float)var + EPSF);
    float bias  = beta[c] - (float)mean * scale;
    coef[c]      = scale;
    coef[64 + c] = bias;
}

__global__ void bn_apply(float* __restrict__ x, const int* __restrict__ cnts,
                         int ci, const float* __restrict__ coef) {
    size_t i = (size_t)blockIdx.x * 256 + threadIdx.x;
    size_t total = (size_t)cnts[ci] * 64;
    if (i < total) {
        int c = (int)(i & 63);
        x[i] = fmaxf(fmaf(x[i], coef[c], coef[64 + c]), 0.0f);
    }
}

// ---------------- final sparse -> dense scatter ----------------
// out layout: (B, CM*D2, H, W): out[((b*64+c)*D2 + z)*H*W + y*W + x]
__global__ void scatter_dense(const float* __restrict__ x2,
                              const int* __restrict__ coords2,
                              const int* __restrict__ cnts,
                              float* __restrict__ out) {
    size_t i = (size_t)blockIdx.x * 256 + threadIdx.x;
    size_t total = (size_t)cnts[2] * 64;
    if (i >= total) return;
    int row = (int)(i >> 6);
    int c   = (int)(i & 63);
    int b = coords2[row*4+0], z = coords2[row*4+1], y = coords2[row*4+2], x = coords2[row*4+3];
    out[(((size_t)(b * 64 + c) * D2C + z) * HH + y) * WW + x] = x2[i];
}

// ---------------- host orchestration ----------------
static inline size_t align_up(size_t v, size_t a) { return (v + a - 1) & ~(a - 1); }

extern "C" void kernel_launch(void* const* d_in, const int* in_sizes, int n_in,
                              void* d_out, int out_size, void* d_ws, size_t ws_size,
                              hipStream_t stream) {
    (void)in_sizes; (void)n_in; (void)ws_size;

    const float* vf       = (const float*)d_in[0];   // [N][128]
    const int*   coors    = (const int*)  d_in[1];   // [N][4]
    const float* w_subm0  = (const float*)d_in[3];   // [27][128][64]
    const float* g0       = (const float*)d_in[4];
    const float* b0       = (const float*)d_in[5];
    const float* w_s1     = (const float*)d_in[6];   // [3][64][64]
    const float* gs1      = (const float*)d_in[7];
    const float* bs1      = (const float*)d_in[8];
    const float* w_subm1a = (const float*)d_in[9];   // [27][64][64]
    const float* g1a      = (const float*)d_in[10];
    const float* b1a      = (const float*)d_in[11];
    const float* w_subm1b = (const float*)d_in[12];
    const float* g1b      = (const float*)d_in[13];
    const float* b1b      = (const float*)d_in[14];
    const float* w_s2     = (const float*)d_in[15];  // [3][64][64]
    const float* gs2      = (const float*)d_in[16];
    const float* bs2      = (const float*)d_in[17];

    // ---- workspace carve-up ----
    char* ws = (char*)d_ws;
    size_t cur = 0;
    auto take = [&](size_t bytes) -> char* {
        char* p = ws + cur;
        cur = align_up(cur + bytes, 256);
        return p;
    };
    const int KS128 = 128 + 2, KS64 = 64 + 2;
    int*   cnts    = (int*)  take(8 * sizeof(int));     // [0]=n0 [1]=n1 [2]=n2
    int*   idx0    = (int*)  take((size_t)M0C * 4);
    int*   occ1    = (int*)  take((size_t)M1C * 4);
    int*   idx1    = (int*)  take((size_t)M1C * 4);
    int*   occ2    = (int*)  take((size_t)M2C * 4);
    int*   idx2    = (int*)  take((size_t)M2C * 4);
    int*   bsum    = (int*)  take(16384 * 4);
    int*   coords1 = (int*)  take((size_t)CAP1 * 4 * 4);
    int*   coords2 = (int*)  take((size_t)CAP2 * 4 * 4);
    float* x0      = (float*)take((size_t)NV0  * 64 * 4);
    float* x1a     = (float*)take((size_t)CAP1 * 64 * 4);
    float* x1b     = (float*)take((size_t)CAP1 * 64 * 4);
    float* x2      = (float*)take((size_t)CAP2 * 64 * 4);
    float* bnp     = (float*)take(256 * 128 * 4);
    float* bncoef  = (float*)take(128 * 4);
    float* wT0     = (float*)take((size_t)27 * 64 * KS128 * 4);
    float* wT1a    = (float*)take((size_t)27 * 64 * KS64 * 4);
    float* wT1b    = (float*)take((size_t)27 * 64 * KS64 * 4);
    float* wTs1    = (float*)take((size_t)3  * 64 * KS64 * 4);
    float* wTs2    = (float*)take((size_t)3  * 64 * KS64 * 4);

    const int G = 256;  // bn partial blocks
    auto bn = [&](float* x, int ci, const float* gamma, const float* beta, int cap) {
        bn_partial<<<G, 64, 0, stream>>>(x, cnts, ci, bnp);
        bn_finalize<<<1, 64, 0, stream>>>(bnp, gamma, beta, cnts, ci, bncoef, G);
        int gr = (cap * 64 + 255) / 256;
        bn_apply<<<gr, 256, 0, stream>>>(x, cnts, ci, bncoef);
    };
    auto xposew = [&](const float* w, float* wT, int taps, int K, int stride) {
        int padn = taps * 64 * stride;
        fill_f32<<<(padn + 255) / 256, 256, 0, stream>>>(wT, 0.0f, padn);
        int tot = taps * K * 64;
        transpose_w<<<(tot + 255) / 256, 256, 0, stream>>>(w, wT, taps, K, stride);
    };

    // ---- counts, level-0 index map, transposed weight panels ----
    init_cnts<<<1, 32, 0, stream>>>(cnts, NV0);
    fill_i32<<<(M0C + 255) / 256, 256, 0, stream>>>(idx0, -1, M0C);
    scatter_idx0<<<(NV0 + 255) / 256, 256, 0, stream>>>(coors, NV0, idx0);
    xposew(w_subm0,  wT0,  27, 128, KS128);
    xposew(w_subm1a, wT1a, 27, 64,  KS64);
    xposew(w_subm1b, wT1b, 27, 64,  KS64);
    xposew(w_s1,     wTs1, 3,  64,  KS64);
    xposew(w_s2,     wTs2, 3,  64,  KS64);

    // ---- subm conv 0: 128 -> 64, 27 taps ----
    conv_wmma<128, 0><<<(NV0 + 63) / 64, 128, 0, stream>>>(
        vf, coors, idx0, wT0, x0, cnts, 0, D0C);
    bn(x0, 0, g0, b0, NV0);

    // ---- sparse conv z #1 (stride 2): occupancy -> scan -> compact ----
    fill_i32<<<(M1C + 255) / 256, 256, 0, stream>>>(occ1, 0, M1C);
    mark_occ<<<(NV0 + 255) / 256, 256, 0, stream>>>(coors, cnts, 0, occ1, D1C);
    {
        int nb = M1C / 256;  // 5500 exactly
        scan_local<<<nb, 256, 0, stream>>>(occ1, idx1, bsum, M1C);
        scan_blocksums<<<1, 256, 0, stream>>>(bsum, nb, cnts + 1);
        scan_finalize<<<nb, 256, 0, stream>>>(occ1, idx1, bsum, coords1, D1C, M1C);
    }
    conv_wmma<64, 1><<<(CAP1 + 63) / 64, 128, 0, stream>>>(
        x0, coords1, idx0, wTs1, x1a, cnts, 1, D0C);
    bn(x1a, 1, gs1, bs1, CAP1);

    // ---- subm conv 1a / 1b: 64 -> 64, 27 taps ----
    conv_wmma<64, 0><<<(CAP1 + 63) / 64, 128, 0, stream>>>(
        x1a, coords1, idx1, wT1a, x1b, cnts, 1, D1C);
    bn(x1b, 1, g1a, b1a, CAP1);
    conv_wmma<64, 0><<<(CAP1 + 63) / 64, 128, 0, stream>>>(
        x1b, coords1, idx1, wT1b, x1a, cnts, 1, D1C);
    bn(x1a, 1, g1b, b1b, CAP1);

    // ---- sparse conv z #2 (stride 2) ----
    fill_i32<<<(M2C + 255) / 256, 256, 0, stream>>>(occ2, 0, M2C);
    mark_occ<<<(CAP1 + 255) / 256, 256, 0, stream>>>(coords1, cnts, 1, occ2, D2C);
    {
        int nb = M2C / 256;  // 2475 exactly
        scan_local<<<nb, 256, 0, stream>>>(occ2, idx2, bsum, M2C);
        scan_blocksums<<<1, 256, 0, stream>>>(bsum, nb, cnts + 2);
        scan_finalize<<<nb, 256, 0, stream>>>(occ2, idx2, bsum, coords2, D2C, M2C);
    }
    conv_wmma<64, 1><<<(CAP2 + 63) / 64, 128, 0, stream>>>(
        x1a, coords2, idx1, wTs2, x2, cnts, 2, D1C);
    bn(x2, 2, gs2, bs2, CAP2);

    // ---- dense output ----
    fill_f32<<<(out_size + 255) / 256, 256, 0, stream>>>((float*)d_out, 0.0f, out_size);
    scatter_dense<<<(CAP2 * 64 + 255) / 256, 256, 0, stream>>>(x2, coords2, cnts, (float*)d_out);
}